// GCN_40973988004062
// MI455X (gfx1250) — compile-verified
//
#include <hip/hip_runtime.h>

#define NNODES 100000
#define NEDGES 1600000
#define NFEAT 128
#define NHID 64
#define NCLASS 16
#define MTILES (NNODES / 16)   // 6250, exact

typedef float v2f __attribute__((ext_vector_type(2)));
typedef float v8f __attribute__((ext_vector_type(8)));

// Non-returning device-scope fp32 atomic add -> global_atomic_add_f32 (no RTN).
__device__ __forceinline__ void atomic_add_f32(float* p, float v) {
    __hip_atomic_fetch_add(p, v, __ATOMIC_RELAXED, __HIP_MEMORY_SCOPE_AGENT);
}

// ---------------------------------------------------------------- init kernels
__global__ __launch_bounds__(256) void gcn_zero_kernel(float4* __restrict__ p, int n4) {
    int i = blockIdx.x * 256 + threadIdx.x;
    if (i < n4) p[i] = make_float4(0.f, 0.f, 0.f, 0.f);
}

__global__ __launch_bounds__(256) void gcn_init_logits_kernel(const float* __restrict__ b2,
                                                              float* __restrict__ out, int total) {
    int i = blockIdx.x * 256 + threadIdx.x;
    if (i < total) out[i] = b2[i & (NCLASS - 1)];
}

// ---------------------------------------------------------------- GEMM1: support1 = x @ W1
// block = 128 threads (4 waves). Each block: one 16-row M tile; wave w does N-tile w*16.
__global__ __launch_bounds__(128) void gcn_gemm1_kernel(const float* __restrict__ x,
                                                        const float* __restrict__ W1,
                                                        float* __restrict__ out) {
    __shared__ float As[16][NFEAT + 4];         // 16 x 132, pad -> conflict-free column reads
    __shared__ float WsT[NHID][NFEAT + 4];      // W1 transposed: WsT[n][k], row stride 132

    const int tid = threadIdx.x;
    const int tileM = blockIdx.x;

    // stage W1 (transposed) : 8192 floats
    for (int i = tid; i < NFEAT * NHID; i += 128) {
        int k = i >> 6, n = i & (NHID - 1);
        WsT[n][k] = W1[i];
    }
    // stage A tile: 16 x 128 floats (coalesced)
    const float* xrow = x + (size_t)tileM * 16 * NFEAT;
    for (int i = tid; i < 16 * NFEAT; i += 128) {
        As[i >> 7][i & (NFEAT - 1)] = xrow[i];
    }
    __syncthreads();

    const int lane = tid & 31;
    const int wave = tid >> 5;          // 0..3  -> N tile
    const int m  = lane & 15;
    const int n  = lane & 15;
    const int kh = (lane >> 4) * 2;     // 0 or 2 (K sub-slice per lane half)
    const int n0 = wave * 16;

    v8f acc = {};
    #pragma unroll 4
    for (int kk = 0; kk < NFEAT; kk += 4) {
        const int k0 = kk + kh;
        v2f a, b;
        a.x = As[m][k0];
        a.y = As[m][k0 + 1];
        b.x = WsT[n0 + n][k0];
        b.y = WsT[n0 + n][k0 + 1];
        acc = __builtin_amdgcn_wmma_f32_16x16x4_f32(false, a, false, b,
                                                    (short)0, acc, false, false);
    }

    float* orow = out + ((size_t)tileM * 16 + (lane >> 4) * 8) * NHID + n0 + n;
    #pragma unroll
    for (int r = 0; r < 8; ++r)
        orow[(size_t)r * NHID] = acc[r];
}

// ---------------------------------------------------------------- SPMM over 64 features
// thread = (edge, feature): f = gid & 63 -> 64-wide coalesced gather + fp32 global atomics
__global__ __launch_bounds__(256) void gcn_spmm64_kernel(const int* __restrict__ src,
                                                         const int* __restrict__ dst,
                                                         const float* __restrict__ ew,
                                                         const float* __restrict__ sup,
                                                         float* __restrict__ out) {
    long gid = (long)blockIdx.x * 256 + threadIdx.x;
    if (gid >= (long)NEDGES * NHID) return;
    int e = (int)(gid >> 6);
    int f = (int)(gid & (NHID - 1));
    int s = src[e];
    int d = dst[e];
    float w = ew[e];
    float msg = w * sup[(size_t)s * NHID + f];
    atomic_add_f32(out + (size_t)d * NHID + f, msg);
}

// ---------------------------------------------------------------- GEMM2: support2 = relu(agg1+b1) @ W2
// block = 256 threads (8 waves). Each wave: one 16-row M tile, single N tile (NCLASS=16).
__global__ __launch_bounds__(256) void gcn_gemm2_kernel(const float* __restrict__ agg1,
                                                        const float* __restrict__ b1,
                                                        const float* __restrict__ W2,
                                                        float* __restrict__ out) {
    __shared__ float WsT[NCLASS][NHID + 4];     // W2 transposed: WsT[n][k]
    __shared__ float bs[NHID];

    const int tid = threadIdx.x;
    for (int i = tid; i < NHID * NCLASS; i += 256) {
        int k = i >> 4, n = i & (NCLASS - 1);
        WsT[n][k] = W2[i];
    }
    if (tid < NHID) bs[tid] = b1[tid];
    __syncthreads();

    const int lane = tid & 31;
    const int wave = tid >> 5;
    const int tileM = blockIdx.x * 8 + wave;
    if (tileM >= MTILES) return;                // whole wave exits together (EXEC stays full)

    const int m  = lane & 15;
    const int n  = lane & 15;
    const int kh = (lane >> 4) * 2;

    const float* arow = agg1 + ((size_t)tileM * 16 + m) * NHID;

    v8f acc = {};
    #pragma unroll 4
    for (int kk = 0; kk < NHID; kk += 4) {
        const int k0 = kk + kh;
        v2f a, b;
        a.x = fmaxf(arow[k0]     + bs[k0],     0.f);   // fused bias + relu
        a.y = fmaxf(arow[k0 + 1] + bs[k0 + 1], 0.f);
        b.x = WsT[n][k0];
        b.y = WsT[n][k0 + 1];
        acc = __builtin_amdgcn_wmma_f32_16x16x4_f32(false, a, false, b,
                                                    (short)0, acc, false, false);
    }

    float* orow = out + ((size_t)tileM * 16 + (lane >> 4) * 8) * NCLASS + n;
    #pragma unroll
    for (int r = 0; r < 8; ++r)
        orow[(size_t)r * NCLASS] = acc[r];
}

// ---------------------------------------------------------------- SPMM over 16 features
__global__ __launch_bounds__(256) void gcn_spmm16_kernel(const int* __restrict__ src,
                                                         const int* __restrict__ dst,
                                                         const float* __restrict__ ew,
                                                         const float* __restrict__ sup,
                                                         float* __restrict__ out) {
    long gid = (long)blockIdx.x * 256 + threadIdx.x;
    if (gid >= (long)NEDGES * NCLASS) return;
    int e = (int)(gid >> 4);
    int f = (int)(gid & (NCLASS - 1));
    int s = src[e];
    int d = dst[e];
    float w = ew[e];
    float msg = w * sup[(size_t)s * NCLASS + f];
    atomic_add_f32(out + (size_t)d * NCLASS + f, msg);
}

// ---------------------------------------------------------------- log_softmax in place (16 classes)
__global__ __launch_bounds__(256) void gcn_logsoftmax_kernel(float* __restrict__ logits) {
    int node = blockIdx.x * 256 + threadIdx.x;
    if (node >= NNODES) return;
    float4* p = (float4*)(logits + (size_t)node * NCLASS);
    float4 v[4];
    #pragma unroll
    for (int i = 0; i < 4; ++i) v[i] = p[i];
    float* f = (float*)v;
    float mx = f[0];
    #pragma unroll
    for (int i = 1; i < 16; ++i) mx = fmaxf(mx, f[i]);
    float sum = 0.f;
    #pragma unroll
    for (int i = 0; i < 16; ++i) sum += __expf(f[i] - mx);
    float lse = mx + __logf(sum);
    #pragma unroll
    for (int i = 0; i < 16; ++i) f[i] -= lse;
    #pragma unroll
    for (int i = 0; i < 4; ++i) p[i] = v[i];
}

// ---------------------------------------------------------------- launch
extern "C" void kernel_launch(void* const* d_in, const int* in_sizes, int n_in,
                              void* d_out, int out_size, void* d_ws, size_t ws_size,
                              hipStream_t stream) {
    const float* x   = (const float*)d_in[0];
    const int*   src = (const int*)  d_in[1];
    const int*   dst = (const int*)  d_in[2];
    const float* ew  = (const float*)d_in[3];
    const float* W1  = (const float*)d_in[4];
    const float* b1  = (const float*)d_in[5];
    const float* W2  = (const float*)d_in[6];
    const float* b2  = (const float*)d_in[7];
    float* out = (float*)d_out;

    float* ws       = (float*)d_ws;
    float* support1 = ws;                                   // NNODES*NHID   = 6.4M floats
    float* agg1     = ws + (size_t)NNODES * NHID;           // NNODES*NHID   = 6.4M floats
    float* support2 = ws + 2 * (size_t)NNODES * NHID;       // NNODES*NCLASS = 1.6M floats

    // zero agg1 (atomic accumulator), seed d_out logits with b2
    {
        int n4 = NNODES * NHID / 4;
        gcn_zero_kernel<<<(n4 + 255) / 256, 256, 0, stream>>>((float4*)agg1, n4);
        int nl = NNODES * NCLASS;
        gcn_init_logits_kernel<<<(nl + 255) / 256, 256, 0, stream>>>(b2, out, nl);
    }

    // layer 1
    gcn_gemm1_kernel<<<MTILES, 128, 0, stream>>>(x, W1, support1);
    {
        long total = (long)NEDGES * NHID;
        int blocks = (int)((total + 255) / 256);
        gcn_spmm64_kernel<<<blocks, 256, 0, stream>>>(src, dst, ew, support1, agg1);
    }

    // layer 2 (bias+relu fused into GEMM2 operand load)
    gcn_gemm2_kernel<<<(MTILES + 7) / 8, 256, 0, stream>>>(agg1, b1, W2, support2);
    {
        long total = (long)NEDGES * NCLASS;
        int blocks = (int)((total + 255) / 256);
        gcn_spmm16_kernel<<<blocks, 256, 0, stream>>>(src, dst, ew, support2, out);
    }

    // log_softmax in place on d_out
    gcn_logsoftmax_kernel<<<(NNODES + 255) / 256, 256, 0, stream>>>(out);
}